// PositionLoss_8452495638692
// MI455X (gfx1250) — compile-verified
//
#include <hip/hip_runtime.h>
#include <hip/hip_bf16.h>

typedef __attribute__((ext_vector_type(2))) float v2f;
typedef __attribute__((ext_vector_type(8))) float v8f;

#define B_DIM   8
#define H_DIM   320
#define W_DIM   448
#define HW      (H_DIM * W_DIM)          // 143360
#define N_OFF   9
#define N_FLOW  4
#define NPIX    (B_DIM * HW)             // 1,146,880
#define BLOCK   256
#define NBLK    (NPIX / BLOCK)           // 4480 (exact)

// ---------------------------------------------------------------------------
// Wave32 sum via V_WMMA_F32_16X16X4_F32 with an all-ones B matrix.
// A layout (16x4 f32, 2 VGPRs/lane): lanes 0-15 hold (M=lane, K=0/1),
// lanes 16-31 hold (M=lane-16, K=2/3). Putting the partial in a.x and 0 in a.y
// gives A[m][0]=p[m], A[m][2]=p[m+16]. With B=ones, D[m][n]=p[m]+p[m+16].
// Each lane sums its 8 D VGPRs (rows 0-7 in lanes 0-15, rows 8-15 in lanes
// 16-31); one shfl_xor(16) completes the full-wave sum on every lane.
// Requires EXEC == all 1s (call from full, non-divergent waves only).
// ---------------------------------------------------------------------------
__device__ __forceinline__ float wave_sum_wmma(float p) {
    v2f a; a.x = p;    a.y = 0.0f;
    v2f b; b.x = 1.0f; b.y = 1.0f;
    v8f c = {};
    c = __builtin_amdgcn_wmma_f32_16x16x4_f32(
        /*neg_a=*/false, a, /*neg_b=*/false, b,
        /*c_mod=*/(short)0, c, /*reuse_a=*/false, /*reuse_b=*/false);
    float s = c[0] + c[1] + c[2] + c[3] + c[4] + c[5] + c[6] + c[7];
    s += __shfl_xor(s, 16, 32);   // wave32: pair (L, L+16) holds the two halves
    return s;
}

// Deterministic block sum: WMMA wave reduce, then serial sum of the 8 wave
// totals by thread 0 (fixed order -> bitwise deterministic).
__device__ __forceinline__ float block_sum(float p) {
    __shared__ float sw[BLOCK / 32];
    float s = wave_sum_wmma(p);
    const int lane = threadIdx.x & 31;
    const int wid  = threadIdx.x >> 5;
    if (lane == 0) sw[wid] = s;
    __syncthreads();
    float t = 0.0f;
    if (threadIdx.x == 0) {
        for (int i = 0; i < BLOCK / 32; ++i) t += sw[i];
    }
    return t;   // valid on thread 0 only
}

__global__ void pos_loss_main(const float* __restrict__ offset,
                              const float* __restrict__ flow,
                              float* __restrict__ partial) {
    const int tid = blockIdx.x * BLOCK + threadIdx.x;   // < NPIX exactly
    const int hw  = tid % HW;                            // h*W + w
    const int b   = tid / HW;

    // ---- per-pixel flow channels: u[j]=f[j], v[j]=f[j+1], j=0..3 ----
    const float* fb = flow + (size_t)b * (2 * N_FLOW) * HW + hw;
    float f[N_FLOW + 1];
#pragma unroll
    for (int j = 0; j <= N_FLOW; ++j) f[j] = fb[(size_t)j * HW];

    float u[N_FLOW], v[N_FLOW], invd[N_FLOW], invsq[N_FLOW], umin[N_FLOW], umax[N_FLOW];
#pragma unroll
    for (int j = 0; j < N_FLOW; ++j) {
        u[j] = f[j];
        v[j] = f[j + 1];
        const float den = u[j] * u[j] + v[j] * v[j];
        invd[j]  = 1.0f / den;           // hoisted: 4 rcp instead of 36 divs
        invsq[j] = 1.0f / sqrtf(den);    // hoisted: 4 rsqrt-equiv instead of 36
        umin[j]  = fminf(0.0f, u[j]);
        umax[j]  = fmaxf(0.0f, u[j]);
    }

    const float* ob = offset + (size_t)b * (2 * N_OFF) * HW + hw;
    float acc = 0.0f;
#pragma unroll
    for (int i = 0; i < N_OFF; ++i) {
        if (i + 1 < N_OFF) {   // gfx1250 global_prefetch_b8 for next channel pair
            __builtin_prefetch(ob + (size_t)(i + 1) * HW, 0, 0);
            __builtin_prefetch(ob + (size_t)(i + 1 + N_OFF) * HW, 0, 0);
        }
        const float x  = ob[(size_t)i * HW];
        const float y  = ob[(size_t)(i + N_OFF) * HW];
        const float d1 = sqrtf(x * x + y * y);
        float best = __builtin_inff();
#pragma unroll
        for (int j = 0; j < N_FLOW; ++j) {
            const float uu = u[j], vv = v[j];
            // foot of perpendicular (as written in the torch/jax code)
            const float x0  = (uu * x * x + uu * vv * y) * invd[j];
            const float gef = (x0 >= umin[j]) ? 1.0f : 0.0f;  // NaN -> 0
            const float lef = (x0 <= umax[j]) ? 1.0f : 0.0f;
            // point-to-line distance; inf -> 0, then float-mask multiply
            float md1 = fabsf(vv * x - uu * y) * invsq[j];
            if (md1 == __builtin_inff()) md1 = 0.0f;
            md1 = md1 * gef * lef;                 // keeps NaN semantics of ref
            // endpoint fallback
            const float dx = x - uu, dy = y - vv;
            const float d2  = sqrtf(dx * dx + dy * dy);
            const float md2 = fminf(d1, d2);
            const float md  = (md1 == 0.0f) ? md2 : md1;
            best = fminf(best, md);
        }
        acc += best;
    }

    const float bs = block_sum(acc);
    if (threadIdx.x == 0) partial[blockIdx.x] = bs;
}

__global__ void pos_loss_reduce(const float* __restrict__ partial, int n,
                                float* __restrict__ out) {
    float p = 0.0f;
    for (int i = threadIdx.x; i < n; i += BLOCK) p += partial[i];  // fixed order
    const float t = block_sum(p);
    if (threadIdx.x == 0) {
        // loss = sum / n_off / (H*W)
        out[0] = t * (1.0f / (float)(N_OFF * H_DIM * W_DIM));
    }
}

extern "C" void kernel_launch(void* const* d_in, const int* in_sizes, int n_in,
                              void* d_out, int out_size, void* d_ws, size_t ws_size,
                              hipStream_t stream) {
    const float* offset = (const float*)d_in[0];   // [8,18,320,448] f32
    const float* flow   = (const float*)d_in[1];   // [8, 8,320,448] f32
    float*       out    = (float*)d_out;           // scalar f32
    float*       part   = (float*)d_ws;            // NBLK floats of scratch

    pos_loss_main  <<<NBLK, BLOCK, 0, stream>>>(offset, flow, part);
    pos_loss_reduce<<<1,    BLOCK, 0, stream>>>(part, NBLK, out);
}